// CausalSelfAttention3_44169443672370
// MI455X (gfx1250) — compile-verified
//
#include <hip/hip_runtime.h>
#include <math.h>

typedef __attribute__((ext_vector_type(2))) float v2f;
typedef __attribute__((ext_vector_type(8))) float v8f;
typedef __attribute__((ext_vector_type(4))) unsigned int u32x4;
typedef __attribute__((ext_vector_type(8))) unsigned int u32x8;

#define T_SZ   64
#define DIM    128
#define NH     4
#define HS     32
#define NG     4
#define GT     16
#define MROWS  17    // merged rows per (head,group) pair
#define MPAD   33    // merged row stride in floats (odd -> bank spread)
#define XPAD   132   // x/o row stride in floats (132 % 64 == 4 -> conflict-free col reads)

__global__ __launch_bounds__(256) void fused_block_attn_kernel(
    const float* __restrict__ x,   // [B,64,128]
    const float* __restrict__ y0,  // [B,4,4,1,32]
    const float* __restrict__ y1,
    const float* __restrict__ y2,
    const float* __restrict__ Wa,  // [128,384]
    const float* __restrict__ Wp,  // [128,128]
    float* __restrict__ out)       // [B,64,128]
{
    __shared__ float xo[T_SZ * XPAD];          // x staging (TDM target), later reused for o
    __shared__ float mq[16 * MROWS * MPAD];    // merged q  [pair][17][32 (+pad)]
    __shared__ float mk[16 * MROWS * MPAD];
    __shared__ float mv[16 * MROWS * MPAD];

    const int tid  = threadIdx.x;
    const int lane = tid & 31;
    const int wave = tid >> 5;
    const int b    = blockIdx.x;

    // ---------------- stage x[b] -> LDS via Tensor Data Mover ---------------
    // 2D tile: 128 (dim0) x 64 rows (dim1), fp32, LDS row padding 4 DWORDs
    // after every 128 DWORDs -> row stride 132 floats == XPAD.
    if (wave == 0) {
        unsigned long long ga =
            (unsigned long long)(const void*)(x + (size_t)b * T_SZ * DIM);
        unsigned int ldsoff = (unsigned int)(unsigned long long)(const void*)&xo[0];
        u32x4 g0;
        g0[0] = 1u;                                   // count=1, no gather
        g0[1] = ldsoff;                               // lds_addr
        g0[2] = (unsigned int)ga;                     // global_addr[31:0]
        g0[3] = (unsigned int)(ga >> 32) | 0x80000000u; // addr[56:32] | type=2
        u32x8 g1;
        g1[0] = (2u << 16)        // data_size = 4B
              | (1u << 20)        // pad_enable
              | (6u << 22)        // pad_interval code 6 -> every 128 DWORDs
              | (3u << 25);       // pad_amount  code 3 -> 4 DWORDs
        g1[1] = (unsigned int)DIM << 16;   // tensor_dim0 = 128 (bits 79:48, low part)
        g1[2] = (unsigned int)T_SZ << 16;  // tensor_dim1 = 64  (bits 111:80, low part)
        g1[3] = (unsigned int)DIM << 16;   // tile_dim0 = 128   (bits 127:112)
        g1[4] = (unsigned int)T_SZ;        // tile_dim1 = 64, tile_dim2 = 0
        g1[5] = (unsigned int)DIM;         // tensor_dim0_stride = 128
        g1[6] = 0u;
        g1[7] = 0u;
        asm volatile("tensor_load_to_lds %0, %1" :: "s"(g0), "s"(g1) : "memory");
        __builtin_amdgcn_s_wait_tensorcnt(0);
    }

    // ---------------- fill summary slots of merged q/k/v from y0/y1/y2 ------
    // group 0: slot 16 <- y[b,h,0]; group g>0: slot 0 <- y[b,h,g-1]
    for (int idx = tid; idx < 3 * 16 * HS; idx += 256) {
        int part = idx >> 9;             // 0:q 1:k 2:v  (512 elems each)
        int rem  = idx & 511;
        int p    = rem >> 5;             // pair = h*4+g
        int hs   = rem & 31;
        int g    = p & 3;
        int h    = p >> 2;
        const float* y   = (part == 0) ? y0 : (part == 1) ? y1 : y2;
        float*       buf = (part == 0) ? mq : (part == 1) ? mk : mv;
        int srcg = (g == 0) ? 0 : (g - 1);
        int slot = (g == 0) ? 16 : 0;
        buf[(p * MROWS + slot) * MPAD + hs] =
            y[(((size_t)b * NH + h) * NG + srcg) * HS + hs];
    }
    __syncthreads();

    // ---------------- GEMM1: qkv = x @ W_attn, scatter into merged layout ---
    // 24 N-tiles of 16 -> 3 per wave; 4 M-tiles accumulated per nt.
    // B fragments software-pipelined one K-step ahead.
    {
        const int nlow    = lane & 15;
        const int kb_lane = 2 * (lane >> 4);
        const int nbase   = wave * 48;
        for (int nt = 0; nt < 3; ++nt) {
            const int ncol = nbase + nt * 16 + nlow;
            v8f acc[4];
            #pragma unroll
            for (int mt = 0; mt < 4; ++mt)
                acc[mt] = (v8f){0.f,0.f,0.f,0.f,0.f,0.f,0.f,0.f};

            v2f bf;
            bf.x = Wa[(size_t)kb_lane * 384 + ncol];
            bf.y = Wa[(size_t)(kb_lane + 1) * 384 + ncol];
            for (int k0 = 0; k0 < DIM; k0 += 4) {
                const int kbn = (((k0 + 4) & (DIM - 1)) + kb_lane);
                v2f bfn;
                bfn.x = Wa[(size_t)kbn * 384 + ncol];
                bfn.y = Wa[(size_t)(kbn + 1) * 384 + ncol];
                const int kb = k0 + kb_lane;
                #pragma unroll
                for (int mt = 0; mt < 4; ++mt) {
                    v2f af = *(const v2f*)&xo[(mt * 16 + nlow) * XPAD + kb];
                    acc[mt] = __builtin_amdgcn_wmma_f32_16x16x4_f32(
                        false, af, false, bf, (short)0, acc[mt], false, false);
                }
                bf = bfn;
            }

            // scatter C tiles into merged q/k/v (block-merge applied here)
            const int c    = nbase + nt * 16 + nlow;   // 0..383
            const int part = c >> 7;                   // wave-uniform per nt
            const int ch   = c & 127;
            const int h    = ch >> 5;
            const int hs   = ch & 31;
            float* buf = (part == 0) ? mq : (part == 1) ? mk : mv;
            #pragma unroll
            for (int mt = 0; mt < 4; ++mt) {
                #pragma unroll
                for (int r = 0; r < 8; ++r) {
                    int m    = mt * 16 + r + 8 * (lane >> 4);  // token 0..63
                    int g    = m >> 4;
                    int i    = m & 15;
                    int mrow = (g == 0) ? i : (i + 1);
                    buf[((h * NG + g) * MROWS + mrow) * MPAD + hs] = acc[mt][r];
                }
            }
        }
    }
    __syncthreads();

    // ---------------- causal attention per (pair,row), online softmax -------
    {
        const float scale = 0.17677669529663687f;  // 1/sqrt(32)
        for (int t = tid; t < 16 * MROWS; t += 256) {
            int p = t / MROWS;
            int r = t - p * MROWS;
            int g = p & 3;
            int h = p >> 2;
            const float* qrow = &mq[(p * MROWS + r) * MPAD];
            float qv[HS], oacc[HS];
            #pragma unroll
            for (int c = 0; c < HS; ++c) { qv[c] = qrow[c]; oacc[c] = 0.f; }
            float mx = -INFINITY, sum = 0.f;
            for (int j = 0; j <= r; ++j) {
                const float* krow = &mk[(p * MROWS + j) * MPAD];
                float s = 0.f;
                #pragma unroll
                for (int c = 0; c < HS; ++c) s += qv[c] * krow[c];
                s *= scale;
                float nm   = fmaxf(mx, s);
                float corr = __expf(mx - nm);   // first iter: exp(-inf)=0
                float w    = __expf(s - nm);
                sum = sum * corr + w;
                const float* vrow = &mv[(p * MROWS + j) * MPAD];
                #pragma unroll
                for (int c = 0; c < HS; ++c) oacc[c] = oacc[c] * corr + w * vrow[c];
                mx = nm;
            }
            float inv = 1.f / sum;
            bool keep = (g == 0) ? (r < GT) : (r > 0);   // un-merge
            if (keep) {
                int token = g * GT + r - ((g > 0) ? 1 : 0);
                float* orow = &xo[token * XPAD + h * HS];
                #pragma unroll
                for (int c = 0; c < HS; ++c) orow[c] = oacc[c] * inv;
            }
        }
    }
    __syncthreads();

    // ---------------- GEMM2: out = o @ W_proj -------------------------------
    // 8 N-tiles -> 1 per wave; 4 M-tiles each; pipelined B fragments.
    {
        const int nlow    = lane & 15;
        const int kb_lane = 2 * (lane >> 4);
        const int ncol    = wave * 16 + nlow;
        v8f acc[4];
        #pragma unroll
        for (int mt = 0; mt < 4; ++mt)
            acc[mt] = (v8f){0.f,0.f,0.f,0.f,0.f,0.f,0.f,0.f};

        v2f bf;
        bf.x = Wp[(size_t)kb_lane * DIM + ncol];
        bf.y = Wp[(size_t)(kb_lane + 1) * DIM + ncol];
        for (int k0 = 0; k0 < DIM; k0 += 4) {
            const int kbn = (((k0 + 4) & (DIM - 1)) + kb_lane);
            v2f bfn;
            bfn.x = Wp[(size_t)kbn * DIM + ncol];
            bfn.y = Wp[(size_t)(kbn + 1) * DIM + ncol];
            const int kb = k0 + kb_lane;
            #pragma unroll
            for (int mt = 0; mt < 4; ++mt) {
                v2f af = *(const v2f*)&xo[(mt * 16 + nlow) * XPAD + kb];
                acc[mt] = __builtin_amdgcn_wmma_f32_16x16x4_f32(
                    false, af, false, bf, (short)0, acc[mt], false, false);
            }
            bf = bfn;
        }
        float* ob = out + (size_t)b * T_SZ * DIM;
        #pragma unroll
        for (int mt = 0; mt < 4; ++mt) {
            #pragma unroll
            for (int r = 0; r < 8; ++r) {
                int row = mt * 16 + r + 8 * (lane >> 4);
                ob[(size_t)row * DIM + ncol] = acc[mt][r];
            }
        }
    }
}

extern "C" void kernel_launch(void* const* d_in, const int* in_sizes, int n_in,
                              void* d_out, int out_size, void* d_ws, size_t ws_size,
                              hipStream_t stream) {
    (void)n_in; (void)d_ws; (void)ws_size; (void)out_size;
    const float* x  = (const float*)d_in[0];
    const float* y0 = (const float*)d_in[1];
    const float* y1 = (const float*)d_in[2];
    const float* y2 = (const float*)d_in[3];
    const float* Wa = (const float*)d_in[4];
    const float* Wp = (const float*)d_in[5];
    float* out = (float*)d_out;
    const int B = in_sizes[0] / (T_SZ * DIM);   // 4096
    fused_block_attn_kernel<<<B, 256, 0, stream>>>(x, y0, y1, y2, Wa, Wp, out);
}